// graphNet_16415365005701
// MI455X (gfx1250) — compile-verified
//
#include <hip/hip_runtime.h>
#include <math.h>

#define NODES 100000
#define EDGES 1600000
#define CCH   21

typedef __attribute__((ext_vector_type(2))) float v2f;
typedef __attribute__((ext_vector_type(8))) float v8f;

#define ATOMIC_ADDF(p, v) unsafeAtomicAdd((p), (v))

static __host__ __device__ inline int roundup4(int x) { return (x + 3) & ~3; }

// ---------------- elementwise / degree kernels ----------------

__global__ __launch_bounds__(256) void fill_zero(float* p, int n) {
  int i = blockIdx.x * 256 + threadIdx.x;
  if (i < n) p[i] = 0.0f;
}

__global__ __launch_bounds__(256) void deg_accum(const int* __restrict__ dst,
                                                 float* __restrict__ deg, int e) {
  int i = blockIdx.x * 256 + threadIdx.x;
  if (i < e) ATOMIC_ADDF(&deg[dst[i]], 1.0f);
}

__global__ __launch_bounds__(256) void deg_to_dinv(float* deg, int n) {
  int i = blockIdx.x * 256 + threadIdx.x;
  if (i < n) deg[i] = rsqrtf(deg[i] + 1.0f);
}

// Zero-pad layer-1 input: XP[N x 36] from x[N x 35]
__global__ __launch_bounds__(256) void pad_x(const float* __restrict__ x,
                                             float* __restrict__ xp) {
  long long idx = (long long)blockIdx.x * 256 + threadIdx.x;
  if (idx >= (long long)NODES * 36) return;
  int node = (int)(idx / 36);
  int k    = (int)(idx % 36);
  xp[idx] = (k < 35) ? x[(size_t)node * 35 + k] : 0.0f;
}

// Pack W[K x Nout] into zero-filled WMMA B fragments:
// Wp[((t*kTiles + kt)*32 + lane)*2 + {0,1}], lane half selects K pair.
__global__ __launch_bounds__(256) void pack_W(const float* __restrict__ W,
                                              float* __restrict__ Wp,
                                              int K, int Nout, int kTiles, int total) {
  int idx = blockIdx.x * 256 + threadIdx.x;   // (t*kTiles + kt)*32 + lane
  if (idx >= total) return;
  int lane = idx & 31;
  int kt   = (idx >> 5) % kTiles;
  int t    = (idx >> 5) / kTiles;
  int half = lane >> 4, l15 = lane & 15;
  int n  = t * 16 + l15;
  int kk = kt * 4 + half * 2;
  float v0 = (kk     < K && n < Nout) ? W[(size_t)kk       * Nout + n] : 0.0f;
  float v1 = (kk + 1 < K && n < Nout) ? W[(size_t)(kk + 1) * Nout + n] : 0.0f;
  Wp[(size_t)idx * 2]     = v0;
  Wp[(size_t)idx * 2 + 1] = v1;
}

// agg = h * dinv^2 + bias for ch<dout, 0 in pad columns (ldo stride)
__global__ __launch_bounds__(256) void init_agg(const float* __restrict__ h,
                                                const float* __restrict__ dinv,
                                                const float* __restrict__ bias,
                                                float* __restrict__ agg,
                                                int dout, int ldo, long long tot) {
  long long idx = (long long)blockIdx.x * 256 + threadIdx.x;
  if (idx >= tot) return;
  int node = (int)(idx / ldo);
  int ch   = (int)(idx % ldo);
  float di = dinv[node];
  agg[idx] = (ch < dout) ? h[idx] * di * di + bias[ch] : 0.0f;
}

__global__ __launch_bounds__(256) void relu_k(float* p, long long n) {
  long long i = (long long)blockIdx.x * 256 + threadIdx.x;
  if (i < n) p[i] = fmaxf(p[i], 0.0f);
}

// ---------------- WMMA fp32 GEMM: H[M x ldo] = X[M x K4] * Wpack ----------------
// One wave32 per 16x16 output tile; unconditional inner loop (operands are
// zero-padded), dual accumulators to break the WMMA C-chain.
__global__ __launch_bounds__(256) void gemm_wmma(const float* __restrict__ X,
                                                 const float* __restrict__ Wp,
                                                 float* __restrict__ H,
                                                 int K4, int kTiles,
                                                 int ldo, int mtiles) {
  int tileM = blockIdx.x * 8 + threadIdx.y;   // wave-uniform
  int tileN = blockIdx.y;
  if (tileM >= mtiles) return;                // whole wave exits together

  int lane = threadIdx.x;        // 0..31
  int half = lane >> 4;
  int l15  = lane & 15;

  int m = tileM * 16 + l15;
  const float* __restrict__ xrow = X + (size_t)m * K4 + half * 2;
  const v2f*   __restrict__ bfrag = (const v2f*)Wp + (size_t)tileN * kTiles * 32 + lane;

  v8f c0 = {0.f, 0.f, 0.f, 0.f, 0.f, 0.f, 0.f, 0.f};
  v8f c1 = {0.f, 0.f, 0.f, 0.f, 0.f, 0.f, 0.f, 0.f};

  int kt = 0;
#pragma unroll 2
  for (; kt + 1 < kTiles; kt += 2) {
    v2f a0 = *(const v2f*)(xrow + (size_t)kt * 4);
    v2f b0 = bfrag[(size_t)kt * 32];
    v2f a1 = *(const v2f*)(xrow + (size_t)kt * 4 + 4);
    v2f b1 = bfrag[(size_t)(kt + 1) * 32];
    c0 = __builtin_amdgcn_wmma_f32_16x16x4_f32(false, a0, false, b0, (short)0, c0,
                                               false, false);
    c1 = __builtin_amdgcn_wmma_f32_16x16x4_f32(false, a1, false, b1, (short)0, c1,
                                               false, false);
  }
  if (kt < kTiles) {
    v2f a0 = *(const v2f*)(xrow + (size_t)kt * 4);
    v2f b0 = bfrag[(size_t)kt * 32];
    c0 = __builtin_amdgcn_wmma_f32_16x16x4_f32(false, a0, false, b0, (short)0, c0,
                                               false, false);
  }
  v8f c = c0 + c1;

  int n = tileN * 16 + l15;
  if (n < ldo) {                              // pad cols (dout..ldo) store exact zeros
    int m0 = tileM * 16 + half * 8;           // VGPR r holds M = m0 + r
    float* hp = H + (size_t)m0 * ldo + n;
#pragma unroll
    for (int r = 0; r < 8; ++r)
      hp[(size_t)r * ldo] = c[r];
  }
}

// ---------------- edge scatter: one wave per edge, lanes stride channels ----------------
__global__ __launch_bounds__(256) void scatter_edges(const int* __restrict__ src,
                                                     const int* __restrict__ dst,
                                                     const float* __restrict__ dinv,
                                                     const float* __restrict__ h,
                                                     float* __restrict__ agg,
                                                     int e, int dout, int ldo) {
  int wave = blockIdx.x * 8 + (threadIdx.x >> 5);
  int lane = threadIdx.x & 31;
  if (wave >= e) return;
  int s = src[wave], d = dst[wave];
  float ne = dinv[s] * dinv[d];
  const float* __restrict__ hs = h + (size_t)s * ldo;
  float* __restrict__ ad = agg + (size_t)d * ldo;
  for (int ch = lane; ch < dout; ch += 32)
    ATOMIC_ADDF(&ad[ch], hs[ch] * ne);
}

// ---------------- VGAE head: reparameterize + log_softmax over 21 channels ----------------
__global__ __launch_bounds__(256) void vgae_head(const float* __restrict__ mu,
                                                 const float* __restrict__ ls,
                                                 const float* __restrict__ eps,
                                                 float* __restrict__ out,
                                                 int ldo, int n_nodes) {
  int t = blockIdx.x * 256 + threadIdx.x;
  if (t >= n_nodes) return;
  const float* mp = mu  + (size_t)t * ldo;
  const float* lp = ls  + (size_t)t * ldo;
  const float* ep = eps + (size_t)t * CCH;
  float z[CCH];
  float mx = -3.0e38f;
#pragma unroll
  for (int j = 0; j < CCH; ++j) {
    float l = fminf(lp[j], 10.0f);
    float zz = mp[j] + ep[j] * expf(l);
    z[j] = zz;
    mx = fmaxf(mx, zz);
  }
  float s = 0.0f;
#pragma unroll
  for (int j = 0; j < CCH; ++j) s += expf(z[j] - mx);
  float lse = mx + logf(s);
  float* op = out + (size_t)t * CCH;
#pragma unroll
  for (int j = 0; j < CCH; ++j) op[j] = z[j] - lse;
}

// ---------------- host-side layer driver ----------------
// in: [NODES x K4in] zero-padded activations (K logical = din, K4in = roundup4)
static void gcn_layer(const float* in, int din, int K4in, int dout,
                      const float* W, const float* b, float* Wpack,
                      float* h, float* agg, const float* dinv,
                      const int* src, const int* dst, hipStream_t stream) {
  const int mtiles = NODES / 16;                 // 6250 exactly
  const int ntiles = (dout + 15) / 16;
  const int kTiles = K4in / 4;
  const int ldo    = roundup4(dout);

  int packTot = ntiles * kTiles * 32;
  pack_W<<<(packTot + 255) / 256, 256, 0, stream>>>(W, Wpack, din, dout, kTiles, packTot);

  dim3 grid((mtiles + 7) / 8, ntiles), block(32, 8);
  gemm_wmma<<<grid, block, 0, stream>>>(in, Wpack, h, K4in, kTiles, ldo, mtiles);

  long long tot = (long long)NODES * ldo;
  init_agg<<<(unsigned)((tot + 255) / 256), 256, 0, stream>>>(h, dinv, b, agg, dout, ldo, tot);
  scatter_edges<<<(EDGES + 7) / 8, 256, 0, stream>>>(src, dst, dinv, h, agg, EDGES, dout, ldo);
}

extern "C" void kernel_launch(void* const* d_in, const int* in_sizes, int n_in,
                              void* d_out, int out_size, void* d_ws, size_t ws_size,
                              hipStream_t stream) {
  const float* x    = (const float*)d_in[0];
  const int*   ei   = (const int*)d_in[1];
  const float* eps  = (const float*)d_in[2];
  const float* Ws[4] = {(const float*)d_in[3], (const float*)d_in[5],
                        (const float*)d_in[7], (const float*)d_in[9]};
  const float* bs[4] = {(const float*)d_in[4], (const float*)d_in[6],
                        (const float*)d_in[8], (const float*)d_in[10]};
  const float* Wmu = (const float*)d_in[11];
  const float* bmu = (const float*)d_in[12];
  const float* Wls = (const float*)d_in[13];
  const float* bls = (const float*)d_in[14];

  const int* src = ei;          // edge_index row 0
  const int* dst = ei + EDGES;  // edge_index row 1

  float* ws    = (float*)d_ws;
  float* dinv  = ws;                                  // N
  float* xp    = dinv + NODES;                        // N * 36 (padded layer-1 input)
  float* wpack = xp + (size_t)NODES * 36;             // 262144 floats (packed B frags)
  float* bufA  = wpack + 262144;                      // N * 336 (GEMM out h)
  float* bufB  = bufA + (size_t)NODES * 336;          // N * 336 (agg / activations)

  // degree -> dinv (fully re-initialized each launch: graph-replay deterministic)
  fill_zero<<<(NODES + 255) / 256, 256, 0, stream>>>(dinv, NODES);
  deg_accum<<<(EDGES + 255) / 256, 256, 0, stream>>>(dst, dinv, EDGES);
  deg_to_dinv<<<(NODES + 255) / 256, 256, 0, stream>>>(dinv, NODES);

  // zero-padded layer-1 input
  {
    long long tot = (long long)NODES * 36;
    pad_x<<<(unsigned)((tot + 255) / 256), 256, 0, stream>>>(x, xp);
  }

  // conv1..conv4 (+ReLU); activations ping through bufB with ldo stride
  const int dims[5] = {35, 16 * CCH, 8 * CCH, 4 * CCH, 2 * CCH};
  const float* cur = xp;
  int curK4 = 36;
  for (int l = 0; l < 4; ++l) {
    int dout = dims[l + 1];
    int ldo  = roundup4(dout);
    gcn_layer(cur, dims[l], curK4, dout, Ws[l], bs[l], wpack,
              bufA, bufB, dinv, src, dst, stream);
    long long tot = (long long)NODES * ldo;
    relu_k<<<(unsigned)((tot + 255) / 256), 256, 0, stream>>>(bufB, tot);
    cur = bufB;
    curK4 = ldo;
  }

  // mu / logstd heads: h4 stays in bufB (ld=44, pad cols exact zeros)
  int ldh = roundup4(CCH);                            // 24
  float* hmu   = bufA;
  float* aggmu = bufA + (size_t)NODES * ldh;
  float* hls   = bufA + (size_t)NODES * 2 * ldh;
  float* aggls = bufA + (size_t)NODES * 3 * ldh;
  gcn_layer(bufB, 2 * CCH, curK4, CCH, Wmu, bmu, wpack, hmu, aggmu, dinv, src, dst, stream);
  gcn_layer(bufB, 2 * CCH, curK4, CCH, Wls, bls, wpack, hls, aggls, dinv, src, dst, stream);

  vgae_head<<<(NODES + 255) / 256, 256, 0, stream>>>(aggmu, aggls, eps,
                                                     (float*)d_out, ldh, NODES);
}